// ConnectLayer_43791486550563
// MI455X (gfx1250) — compile-verified
//
#include <hip/hip_runtime.h>
#include <hip/hip_bf16.h>
#include <math.h>

// ---------------------------------------------------------------------------
// ConnectLayer pipeline for MI455X (gfx1250), fp32 WMMA (V_WMMA_F32_16X16X4_F32)
// GEMM kernels use 2x2 register blocking: one wave32 owns a 32x32 macro-tile
// (4x v8f accumulators), so each K=4 chunk feeds 4 WMMAs from 4 b64 loads.
// ---------------------------------------------------------------------------

#define EH   256
#define DH   1024
#define BN   128
#define TW   32      // WIKI sentences (T)
#define SEN  64
#define HIST 4

typedef float v2f __attribute__((ext_vector_type(2)));
typedef float v8f __attribute__((ext_vector_type(8)));

static __device__ __forceinline__ v8f wmma4(v2f a, v2f b, v8f c) {
  // D(16x16,f32) = A(16x4,f32) * B(4x16,f32) + C
  return __builtin_amdgcn_wmma_f32_16x16x4_f32(false, a, false, b, (short)0, c,
                                               false, false);
}

// ---------------------------------------------------------------------------
// K1: gx = x @ Wih^T + bih for both GRU directions (2x2 blocked).
// Backward direction gathers x_rev[t,b] = x[clip(len[b]-1-t,0,31), b] in the
// A-operand address generation.
// grid = (128, 24, 2), block = 32
// ---------------------------------------------------------------------------
__global__ void k_gx(const float* __restrict__ x,
                     const float* __restrict__ WihF, const float* __restrict__ bihF,
                     const float* __restrict__ WihB, const float* __restrict__ bihB,
                     const int* __restrict__ lens,
                     float* __restrict__ gxF, float* __restrict__ gxB) {
  const int lane = threadIdx.x & 31;
  const int m = lane & 15, g = lane >> 4;
  const int mt = blockIdx.x;            // 0..127 -> 32-row tile of (T*B)
  const int nt = blockIdx.y;            // 0..23  -> 32-col tile of 768
  const int dir = blockIdx.z;
  const int t  = mt >> 2;
  const int b0 = (mt & 3) << 5;
  const int n0 = nt << 5;

  const float* Wih = dir ? WihB : WihF;
  const float* bih = dir ? bihB : bihF;
  float*       gx  = dir ? gxB  : gxF;

  const int bA0 = b0 + m, bA1 = b0 + 16 + m;
  int s0 = t, s1 = t;
  if (dir) {
    s0 = lens[bA0] - 1 - t; s0 = s0 < 0 ? 0 : (s0 > TW - 1 ? TW - 1 : s0);
    s1 = lens[bA1] - 1 - t; s1 = s1 < 0 ? 0 : (s1 > TW - 1 ? TW - 1 : s1);
  }
  const float* ar0 = x + ((size_t)s0 * BN + bA0) * 512;
  const float* ar1 = x + ((size_t)s1 * BN + bA1) * 512;
  const float* br0 = Wih + (size_t)(n0 + m) * 512;
  const float* br1 = Wih + (size_t)(n0 + 16 + m) * 512;
  __builtin_prefetch(br0, 0, 1);
  __builtin_prefetch(br1, 0, 1);

  v8f acc00 = {}, acc01 = {}, acc10 = {}, acc11 = {};
  for (int k = 0; k < 512; k += 4) {
    const int k2 = k + 2 * g;
    v2f a0, a1, bb0, bb1;
    a0.x = ar0[k2];  a0.y = ar0[k2 + 1];
    a1.x = ar1[k2];  a1.y = ar1[k2 + 1];
    bb0.x = br0[k2]; bb0.y = br0[k2 + 1];
    bb1.x = br1[k2]; bb1.y = br1[k2 + 1];
    acc00 = wmma4(a0, bb0, acc00);
    acc01 = wmma4(a0, bb1, acc01);
    acc10 = wmma4(a1, bb0, acc10);
    acc11 = wmma4(a1, bb1, acc11);
  }
  const float bi0 = bih[n0 + m], bi1 = bih[n0 + 16 + m];
#pragma unroll
  for (int v = 0; v < 8; ++v) {
    const int r0 = b0 + v + 8 * g;        // D rows of the two m-tiles
    const int r1 = r0 + 16;
    float* p0 = gx + ((size_t)t * BN + r0) * 768 + n0 + m;
    float* p1 = gx + ((size_t)t * BN + r1) * 768 + n0 + m;
    p0[0]  = acc00[v] + bi0;  p0[16] = acc01[v] + bi1;
    p1[0]  = acc10[v] + bi0;  p1[16] = acc11[v] + bi1;
  }
}

// ---------------------------------------------------------------------------
// K2: full 32-step masked GRU in a single kernel.  One block per direction,
// 1024 threads = 32 waves; hidden state h (128x256 f32 = 128 KB) in LDS.
// Each wave owns 4 gate tiles: computes r/z/n GEMM slices (K=256) from LDS
// (A operand shared by the 3 gate WMMAs), applies the gates, then
// double-barrier h update.
// grid = 2, block = 1024, dynamic LDS = 131072 B
// ---------------------------------------------------------------------------
__global__ void k_gru(const float* __restrict__ gxF, const float* __restrict__ gxB,
                      const float* __restrict__ WhhF, const float* __restrict__ bhhF,
                      const float* __restrict__ WhhB, const float* __restrict__ bhhB,
                      const int* __restrict__ lens,
                      float* __restrict__ outF, float* __restrict__ outB) {
  extern __shared__ float sh[];          // h: [128][256]
  const int dir = blockIdx.x;
  const float* gx  = dir ? gxB  : gxF;
  const float* Whh = dir ? WhhB : WhhF;
  const float* bhh = dir ? bhhB : bhhF;
  float*       outs = dir ? outB : outF;

  const int tid  = threadIdx.x;
  const int lane = tid & 31;
  const int wave = tid >> 5;
  const int m = lane & 15, g = lane >> 4;

  for (int i = tid; i < BN * EH; i += 1024) sh[i] = 0.0f;
  __syncthreads();

  for (int t = 0; t < TW; ++t) {
    v8f hnew[4];
#pragma unroll 1
    for (int it = 0; it < 4; ++it) {
      const int tile = wave + it * 32;   // 0..127
      const int mt = tile >> 4, nt = tile & 15;
      const int b0 = mt << 4, n0 = nt << 4;
      const float* arow = sh  + (size_t)(b0 + m) * EH;
      const float* br = Whh + (size_t)(         n0 + m) * EH;
      const float* bz = Whh + (size_t)(EH     + n0 + m) * EH;
      const float* bn = Whh + (size_t)(2 * EH + n0 + m) * EH;
      v8f ar = {}, az = {}, an = {};
      for (int k = 0; k < EH; k += 4) {
        const int k2 = k + 2 * g;
        v2f a, bb;
        a.x = arow[k2]; a.y = arow[k2 + 1];
        bb.x = br[k2];  bb.y = br[k2 + 1];  ar = wmma4(a, bb, ar);
        bb.x = bz[k2];  bb.y = bz[k2 + 1];  az = wmma4(a, bb, az);
        bb.x = bn[k2];  bb.y = bn[k2 + 1];  an = wmma4(a, bb, an);
      }
      v8f hn8;
#pragma unroll
      for (int v = 0; v < 8; ++v) {
        const int ml = v + 8 * g;
        const int b = b0 + ml;
        const int n = n0 + m;
        const float hprev = sh[b * EH + n];
        const size_t gbase = ((size_t)t * BN + b) * 768 + n;
        const float r  = 1.0f / (1.0f + __expf(-(gx[gbase]          + ar[v] + bhh[n])));
        const float z  = 1.0f / (1.0f + __expf(-(gx[gbase + EH]     + az[v] + bhh[EH + n])));
        const float nn = tanhf(gx[gbase + 2 * EH] + r * (an[v] + bhh[2 * EH + n]));
        const float hh = (1.0f - z) * nn + z * hprev;
        const bool valid = t < lens[b];
        outs[((size_t)t * BN + b) * EH + n] = valid ? hh : 0.0f;
        hn8[v] = valid ? hh : hprev;
      }
      hnew[it] = hn8;
    }
    __syncthreads();
#pragma unroll 1
    for (int it = 0; it < 4; ++it) {
      const int tile = wave + it * 32;
      const int mt = tile >> 4, nt = tile & 15;
      const int b0 = mt << 4, n0 = nt << 4;
#pragma unroll
      for (int v = 0; v < 8; ++v)
        sh[(b0 + v + 8 * g) * EH + n0 + m] = hnew[it][v];
    }
    __syncthreads();
  }
}

// ---------------------------------------------------------------------------
// K3: h2 = concat(out_f, re-reversed+masked out_b)   (T,B,512) elementwise
// ---------------------------------------------------------------------------
__global__ void k_h2(const float* __restrict__ outF, const float* __restrict__ outB,
                     const int* __restrict__ lens, float* __restrict__ h2) {
  const int i = blockIdx.x * 256 + threadIdx.x;      // t*65536 + b*512 + c
  if (i >= TW * BN * 512) return;
  const int c = i & 511;
  const int b = (i >> 9) & 127;
  const int t = i >> 16;
  float val;
  if (c < EH) {
    val = outF[((size_t)t * BN + b) * EH + c];
  } else {
    const int l = lens[b];
    int srct = l - 1 - t;
    srct = srct < 0 ? 0 : (srct > TW - 1 ? TW - 1 : srct);
    val = (t < l) ? outB[((size_t)srct * BN + b) * EH + (c - EH)] : 0.0f;
  }
  h2[i] = val;
}

// ---------------------------------------------------------------------------
// K4: tilde_wiki = sum_h hw[h] * tanh(W_tilde [lw-h2 ; lw*h2] + b_tilde)
// Features built on the fly in the A-operand load, 2x2 blocked.
// grid=(128,16), block=32
// ---------------------------------------------------------------------------
__global__ void k_tilde(const float* __restrict__ h2, const float* __restrict__ lw,
                        const float* __restrict__ Wt, const float* __restrict__ bt,
                        float* __restrict__ tilde) {
  const int lane = threadIdx.x & 31;
  const int m = lane & 15, g = lane >> 4;
  const int t = blockIdx.x >> 2, b0 = (blockIdx.x & 3) << 5, o0 = blockIdx.y << 5;
  const float* hr0 = h2 + ((size_t)t * BN + b0 + m) * 512;
  const float* hr1 = h2 + ((size_t)t * BN + b0 + 16 + m) * 512;
  const float* lw0 = lw + (size_t)(b0 + m) * (512 * HIST);        // [c*4 + h]
  const float* lw1 = lw + (size_t)(b0 + 16 + m) * (512 * HIST);
  const float* w0  = Wt + (size_t)(o0 + m) * 1024;
  const float* w1  = Wt + (size_t)(o0 + 16 + m) * 1024;
  __builtin_prefetch(w0, 0, 1);
  __builtin_prefetch(w1, 0, 1);
  const float hw[4] = {0.5f, 0.25f, 0.15f, 0.1f};

  v8f o00 = {}, o01 = {}, o10 = {}, o11 = {};
  for (int h = 0; h < HIST; ++h) {
    v8f a00 = {}, a01 = {}, a10 = {}, a11 = {};
    for (int k = 0; k < 512; k += 4) {           // first half: lw - h2
      const int k2 = k + 2 * g;
      v2f a0, a1, bb0, bb1;
      a0.x = lw0[k2 * 4 + h]       - hr0[k2];
      a0.y = lw0[(k2 + 1) * 4 + h] - hr0[k2 + 1];
      a1.x = lw1[k2 * 4 + h]       - hr1[k2];
      a1.y = lw1[(k2 + 1) * 4 + h] - hr1[k2 + 1];
      bb0.x = w0[k2]; bb0.y = w0[k2 + 1];
      bb1.x = w1[k2]; bb1.y = w1[k2 + 1];
      a00 = wmma4(a0, bb0, a00);
      a01 = wmma4(a0, bb1, a01);
      a10 = wmma4(a1, bb0, a10);
      a11 = wmma4(a1, bb1, a11);
    }
    for (int k = 0; k < 512; k += 4) {           // second half: lw * h2
      const int k2 = k + 2 * g;
      v2f a0, a1, bb0, bb1;
      a0.x = lw0[k2 * 4 + h]       * hr0[k2];
      a0.y = lw0[(k2 + 1) * 4 + h] * hr0[k2 + 1];
      a1.x = lw1[k2 * 4 + h]       * hr1[k2];
      a1.y = lw1[(k2 + 1) * 4 + h] * hr1[k2 + 1];
      bb0.x = w0[512 + k2]; bb0.y = w0[512 + k2 + 1];
      bb1.x = w1[512 + k2]; bb1.y = w1[512 + k2 + 1];
      a00 = wmma4(a0, bb0, a00);
      a01 = wmma4(a0, bb1, a01);
      a10 = wmma4(a1, bb0, a10);
      a11 = wmma4(a1, bb1, a11);
    }
    const float bs0 = bt[o0 + m], bs1 = bt[o0 + 16 + m];   // per-column bias
    const float w = hw[h];
#pragma unroll
    for (int v = 0; v < 8; ++v) {
      o00[v] += w * tanhf(a00[v] + bs0);
      o01[v] += w * tanhf(a01[v] + bs1);
      o10[v] += w * tanhf(a10[v] + bs0);
      o11[v] += w * tanhf(a11[v] + bs1);
    }
  }
#pragma unroll
  for (int v = 0; v < 8; ++v) {
    const int r0 = b0 + v + 8 * g, r1 = r0 + 16;
    float* p0 = tilde + ((size_t)t * BN + r0) * 512 + o0 + m;
    float* p1 = tilde + ((size_t)t * BN + r1) * 512 + o0 + m;
    p0[0] = o00[v];  p0[16] = o01[v];
    p1[0] = o10[v];  p1[16] = o11[v];
  }
}

// ---------------------------------------------------------------------------
// K5: query = h_n @ W_q^T (2x2 blocked)   grid=(4,16), block=32
// ---------------------------------------------------------------------------
__global__ void k_query(const float* __restrict__ hn, const float* __restrict__ Wq,
                        float* __restrict__ q) {
  const int lane = threadIdx.x & 31;
  const int m = lane & 15, g = lane >> 4;
  const int b0 = blockIdx.x << 5, n0 = blockIdx.y << 5;
  const float* ar0 = hn + (size_t)(b0 + m) * 512;
  const float* ar1 = hn + (size_t)(b0 + 16 + m) * 512;
  const float* br0 = Wq + (size_t)(n0 + m) * 512;
  const float* br1 = Wq + (size_t)(n0 + 16 + m) * 512;
  v8f a00 = {}, a01 = {}, a10 = {}, a11 = {};
  for (int k = 0; k < 512; k += 4) {
    const int k2 = k + 2 * g;
    v2f a0, a1, bb0, bb1;
    a0.x = ar0[k2];  a0.y = ar0[k2 + 1];
    a1.x = ar1[k2];  a1.y = ar1[k2 + 1];
    bb0.x = br0[k2]; bb0.y = br0[k2 + 1];
    bb1.x = br1[k2]; bb1.y = br1[k2 + 1];
    a00 = wmma4(a0, bb0, a00);
    a01 = wmma4(a0, bb1, a01);
    a10 = wmma4(a1, bb0, a10);
    a11 = wmma4(a1, bb1, a11);
  }
#pragma unroll
  for (int v = 0; v < 8; ++v) {
    const int r0 = b0 + v + 8 * g, r1 = r0 + 16;
    q[(size_t)r0 * 512 + n0 + m]      = a00[v];
    q[(size_t)r0 * 512 + n0 + 16 + m] = a01[v];
    q[(size_t)r1 * 512 + n0 + m]      = a10[v];
    q[(size_t)r1 * 512 + n0 + 16 + m] = a11[v];
  }
}

// ---------------------------------------------------------------------------
// K6a: key = [h_n1 ; tilde] @ W_k^T, W_k split at the concat boundary,
// 2x2 blocked.  grid=(128,16), block=32
// ---------------------------------------------------------------------------
__global__ void k_key(const float* __restrict__ h_n1, const float* __restrict__ tilde,
                      const float* __restrict__ Wk, float* __restrict__ key) {
  const int lane = threadIdx.x & 31;
  const int m = lane & 15, g = lane >> 4;
  const int t = blockIdx.x >> 2, b0 = (blockIdx.x & 3) << 5, o0 = blockIdx.y << 5;
  const size_t row0 = (size_t)t * BN + b0 + m;
  const size_t row1 = row0 + 16;
  const float* x10 = h_n1  + row0 * 512;
  const float* x11 = h_n1  + row1 * 512;
  const float* x20 = tilde + row0 * 512;
  const float* x21 = tilde + row1 * 512;
  const float* w0  = Wk + (size_t)(o0 + m) * 1024;
  const float* w1  = Wk + (size_t)(o0 + 16 + m) * 1024;
  __builtin_prefetch(w0, 0, 1);
  __builtin_prefetch(w1, 0, 1);
  v8f a00 = {}, a01 = {}, a10 = {}, a11 = {};
  for (int k = 0; k < 512; k += 4) {
    const int k2 = k + 2 * g;
    v2f a0, a1, bb0, bb1;
    a0.x = x10[k2];  a0.y = x10[k2 + 1];
    a1.x = x11[k2];  a1.y = x11[k2 + 1];
    bb0.x = w0[k2];  bb0.y = w0[k2 + 1];
    bb1.x = w1[k2];  bb1.y = w1[k2 + 1];
    a00 = wmma4(a0, bb0, a00);
    a01 = wmma4(a0, bb1, a01);
    a10 = wmma4(a1, bb0, a10);
    a11 = wmma4(a1, bb1, a11);
  }
  for (int k = 0; k < 512; k += 4) {
    const int k2 = k + 2 * g;
    v2f a0, a1, bb0, bb1;
    a0.x = x20[k2];        a0.y = x20[k2 + 1];
    a1.x = x21[k2];        a1.y = x21[k2 + 1];
    bb0.x = w0[512 + k2];  bb0.y = w0[512 + k2 + 1];
    bb1.x = w1[512 + k2];  bb1.y = w1[512 + k2 + 1];
    a00 = wmma4(a0, bb0, a00);
    a01 = wmma4(a0, bb1, a01);
    a10 = wmma4(a1, bb0, a10);
    a11 = wmma4(a1, bb1, a11);
  }
#pragma unroll
  for (int v = 0; v < 8; ++v) {
    const int r0 = b0 + v + 8 * g, r1 = r0 + 16;
    float* p0 = key + ((size_t)t * BN + r0) * 512 + o0 + m;
    float* p1 = key + ((size_t)t * BN + r1) * 512 + o0 + m;
    p0[0] = a00[v];  p0[16] = a01[v];
    p1[0] = a10[v];  p1[16] = a11[v];
  }
}

// ---------------------------------------------------------------------------
// K6b: beta[b][t] = sum_o tanh(query+key) * v_attn   (one wave per (t,b))
// grid=512, block=256
// ---------------------------------------------------------------------------
__global__ void k_beta(const float* __restrict__ q, const float* __restrict__ key,
                       const float* __restrict__ vat, float* __restrict__ beta) {
  const int r = blockIdx.x * 8 + (threadIdx.x >> 5);   // 0..4095 = t*128+b
  const int lane = threadIdx.x & 31;
  const int t = r >> 7, b = r & 127;
  const float* krow = key + (size_t)r * 512;
  const float* qrow = q   + (size_t)b * 512;
  float s = 0.0f;
  for (int o = lane; o < 512; o += 32)
    s += tanhf(qrow[o] + krow[o]) * vat[o];
  for (int off = 16; off; off >>= 1) s += __shfl_xor(s, off, 32);
  if (lane == 0) beta[b * TW + t] = s;
}

// ---------------------------------------------------------------------------
// K7: masked cross-entropy (ignore_index=0) over beta (B,T). 1 block, 128 thr.
// ---------------------------------------------------------------------------
__global__ void k_loss(const float* __restrict__ beta, const int* __restrict__ label,
                       float* __restrict__ out_loss) {
  __shared__ float snll[BN], smsk[BN];
  const int b = threadIdx.x;
  const float* row = beta + b * TW;
  float mx = row[0];
  for (int t = 1; t < TW; ++t) mx = fmaxf(mx, row[t]);
  float se = 0.0f;
  for (int t = 0; t < TW; ++t) se += __expf(row[t] - mx);
  const int lab = label[b];
  const float lp = row[lab] - mx - __logf(se);
  const float msk = (lab != 0) ? 1.0f : 0.0f;
  snll[b] = -lp * msk;
  smsk[b] = msk;
  __syncthreads();
  for (int s = 64; s; s >>= 1) {
    if (b < s) { snll[b] += snll[b + s]; smsk[b] += smsk[b + s]; }
    __syncthreads();
  }
  if (b == 0) out_loss[0] = snll[0] / fmaxf(smsk[0], 1.0f);
}

// ---------------------------------------------------------------------------
// K8: init_h = [h_n ; wiki_cv] @ W_init^T + b_init (gather fused into A load,
// 2x2 blocked).  grid=(4,32), block=32
// ---------------------------------------------------------------------------
__global__ void k_init(const float* __restrict__ hn, const float* __restrict__ h_n1,
                       const int* __restrict__ label,
                       const float* __restrict__ Wi, const float* __restrict__ bi,
                       float* __restrict__ out_init) {
  const int lane = threadIdx.x & 31;
  const int m = lane & 15, g = lane >> 4;
  const int b0 = blockIdx.x << 5, n0 = blockIdx.y << 5;
  const int bA0 = b0 + m, bA1 = b0 + 16 + m;
  const float* x10 = hn + (size_t)bA0 * 512;
  const float* x11 = hn + (size_t)bA1 * 512;
  const float* x20 = h_n1 + ((size_t)label[bA0] * BN + bA0) * 512;
  const float* x21 = h_n1 + ((size_t)label[bA1] * BN + bA1) * 512;
  const float* w0  = Wi + (size_t)(n0 + m) * 1024;
  const float* w1  = Wi + (size_t)(n0 + 16 + m) * 1024;
  v8f a00 = {}, a01 = {}, a10 = {}, a11 = {};
  for (int k = 0; k < 512; k += 4) {
    const int k2 = k + 2 * g;
    v2f a0, a1, bb0, bb1;
    a0.x = x10[k2];  a0.y = x10[k2 + 1];
    a1.x = x11[k2];  a1.y = x11[k2 + 1];
    bb0.x = w0[k2];  bb0.y = w0[k2 + 1];
    bb1.x = w1[k2];  bb1.y = w1[k2 + 1];
    a00 = wmma4(a0, bb0, a00);
    a01 = wmma4(a0, bb1, a01);
    a10 = wmma4(a1, bb0, a10);
    a11 = wmma4(a1, bb1, a11);
  }
  for (int k = 0; k < 512; k += 4) {
    const int k2 = k + 2 * g;
    v2f a0, a1, bb0, bb1;
    a0.x = x20[k2];        a0.y = x20[k2 + 1];
    a1.x = x21[k2];        a1.y = x21[k2 + 1];
    bb0.x = w0[512 + k2];  bb0.y = w0[512 + k2 + 1];
    bb1.x = w1[512 + k2];  bb1.y = w1[512 + k2 + 1];
    a00 = wmma4(a0, bb0, a00);
    a01 = wmma4(a0, bb1, a01);
    a10 = wmma4(a1, bb0, a10);
    a11 = wmma4(a1, bb1, a11);
  }
  const float bi0 = bi[n0 + m], bi1 = bi[n0 + 16 + m];
#pragma unroll
  for (int v = 0; v < 8; ++v) {
    const int r0 = b0 + v + 8 * g, r1 = r0 + 16;
    out_init[(size_t)r0 * DH + n0 + m]      = a00[v] + bi0;
    out_init[(size_t)r0 * DH + n0 + 16 + m] = a01[v] + bi1;
    out_init[(size_t)r1 * DH + n0 + m]      = a10[v] + bi0;
    out_init[(size_t)r1 * DH + n0 + 16 + m] = a11[v] + bi1;
  }
}

// ---------------------------------------------------------------------------
// K9: output gathers
// ---------------------------------------------------------------------------
__global__ void k_cv(const float* __restrict__ h_n1, const int* __restrict__ label,
                     float* __restrict__ o) {
  const int i = blockIdx.x * 256 + threadIdx.x;        // 65536
  if (i >= BN * 512) return;
  const int b = i >> 9, c = i & 511;
  o[i] = h_n1[((size_t)label[b] * BN + b) * 512 + c];
}

__global__ void k_swh(const float* __restrict__ h1, const int* __restrict__ label,
                      float* __restrict__ o) {
  const size_t i = (size_t)blockIdx.x * 256 + threadIdx.x;   // 4194304
  if (i >= (size_t)BN * SEN * 512) return;
  const int c = (int)(i & 511);
  const int s = (int)((i >> 9) & 63);
  const int b = (int)(i >> 15);
  // h1: (SEN, B, WIKI, 512)
  o[i] = h1[(((size_t)s * BN + b) * TW + label[b]) * 512 + c];
}

__global__ void k_sws(const int* __restrict__ wsen, const int* __restrict__ label,
                      float* __restrict__ o) {
  const int i = blockIdx.x * 256 + threadIdx.x;        // 8192
  if (i >= BN * SEN) return;
  const int b = i >> 6, s = i & 63;
  o[i] = (float)wsen[((size_t)b * TW + label[b]) * SEN + s];
}

// ---------------------------------------------------------------------------
// host launcher
// ---------------------------------------------------------------------------
extern "C" void kernel_launch(void* const* d_in, const int* in_sizes, int n_in,
                              void* d_out, int out_size, void* d_ws, size_t ws_size,
                              hipStream_t stream) {
  const float* h_n1      = (const float*)d_in[0];
  const float* h1        = (const float*)d_in[1];
  const float* h_n       = (const float*)d_in[2];
  const float* last_wiki = (const float*)d_in[3];
  const float* W_init    = (const float*)d_in[4];
  const float* b_init    = (const float*)d_in[5];
  const float* W_tilde   = (const float*)d_in[6];
  const float* b_tilde   = (const float*)d_in[7];
  const float* W_q       = (const float*)d_in[8];
  const float* W_k       = (const float*)d_in[9];
  const float* v_attn    = (const float*)d_in[10];
  const float* Wih_f     = (const float*)d_in[11];
  const float* Whh_f     = (const float*)d_in[12];
  const float* bih_f     = (const float*)d_in[13];
  const float* bhh_f     = (const float*)d_in[14];
  const float* Wih_b     = (const float*)d_in[15];
  const float* Whh_b     = (const float*)d_in[16];
  const float* bih_b     = (const float*)d_in[17];
  const float* bhh_b     = (const float*)d_in[18];
  const int*   wiki_sen  = (const int*)d_in[19];
  const int*   label     = (const int*)d_in[20];
  const int*   lens      = (const int*)d_in[21];

  float* out = (float*)d_out;
  // output layout (flat, return order)
  float* o_loss = out;                       // 1
  float* o_init = out + 1;                   // 128*1024
  float* o_cv   = out + 1 + 131072;          // 128*512
  float* o_swh  = out + 1 + 131072 + 65536;  // 128*64*512
  float* o_sws  = out + 1 + 131072 + 65536 + 4194304; // 128*64

  // workspace layout (floats)
  float* ws   = (float*)d_ws;
  float* gxF  = ws;                          // 32*128*768
  float* gxB  = ws + 3145728;                // 32*128*768
  float* outF = ws + 6291456;                // 32*128*256
  float* outB = ws + 7340032;                // 32*128*256
  float* h2   = ws + 8388608;                // 32*128*512
  float* tld  = ws + 10485760;               // 32*128*512
  float* keyb = ws + 12582912;               // 32*128*512
  float* qry  = ws + 14680064;               // 128*512
  float* beta = ws + 14745600;               // 128*32

  k_gx<<<dim3(128, 24, 2), 32, 0, stream>>>(h_n1, Wih_f, bih_f, Wih_b, bih_b,
                                            lens, gxF, gxB);
  k_gru<<<2, 1024, BN * EH * sizeof(float), stream>>>(gxF, gxB, Whh_f, bhh_f,
                                                      Whh_b, bhh_b, lens, outF, outB);
  k_h2<<<(TW * BN * 512 + 255) / 256, 256, 0, stream>>>(outF, outB, lens, h2);
  k_tilde<<<dim3(128, 16), 32, 0, stream>>>(h2, last_wiki, W_tilde, b_tilde, tld);
  k_query<<<dim3(4, 16), 32, 0, stream>>>(h_n, W_q, qry);
  k_key<<<dim3(128, 16), 32, 0, stream>>>(h_n1, tld, W_k, keyb);
  k_beta<<<512, 256, 0, stream>>>(qry, keyb, v_attn, beta);
  k_loss<<<1, 128, 0, stream>>>(beta, label, o_loss);
  k_init<<<dim3(4, 32), 32, 0, stream>>>(h_n, h_n1, label, W_init, b_init, o_init);
  k_cv<<<(BN * 512 + 255) / 256, 256, 0, stream>>>(h_n1, label, o_cv);
  k_swh<<<(BN * SEN * 512 + 255) / 256, 256, 0, stream>>>(h1, label, o_swh);
  k_sws<<<(BN * SEN + 255) / 256, 256, 0, stream>>>(wiki_sen, label, o_sws);
}